// imaging_layer_47218870452774
// MI455X (gfx1250) — compile-verified
//
#include <hip/hip_runtime.h>
#include <math.h>

// Problem constants (match reference: B, N, S_regions)
#define BATCH      256
#define NMESH      500
#define NREG       10
#define ACC_STRIDE 16   // padded accumulator row: [0..9]=regions, [10]=total

typedef __attribute__((ext_vector_type(2))) float v2f;
typedef __attribute__((ext_vector_type(8))) float v8f;

// ---------------------------------------------------------------- zero ws
__global__ void zero_accum(float* __restrict__ acc) {
    int i = blockIdx.x * blockDim.x + threadIdx.x;
    if (i < BATCH * ACC_STRIDE) acc[i] = 0.0f;
}

// --------------------------------------------------- block reduce (256 thr)
__device__ __forceinline__ float block_reduce_256(float v, float* red) {
    // wave32 tree reduce
    #pragma unroll
    for (int off = 16; off > 0; off >>= 1)
        v += __shfl_down(v, off, 32);
    const int lane = threadIdx.x & 31;
    const int wave = threadIdx.x >> 5;
    if (lane == 0) red[wave] = v;
    __syncthreads();
    float s = 0.0f;
    if (threadIdx.x == 0) {
        #pragma unroll
        for (int w = 0; w < 8; ++w) s += red[w];
    }
    return s; // valid in thread 0 only
}

// ------------------------------------------- total intensity per image (HBM)
// grid = (chunks, BATCH), block = 256.  Streams 512 MB with B128 loads.
__global__ void total_kernel(const float* __restrict__ ur,
                             const float* __restrict__ ui,
                             float* __restrict__ acc) {
    __shared__ float red[8];
    const int  b    = blockIdx.y;
    const long base = (long)b * (NMESH * NMESH);
    const float4* r4 = (const float4*)(ur + base);
    const float4* i4 = (const float4*)(ui + base);
    const int n4 = (NMESH * NMESH) / 4; // 62500 float4 per image
    float a = 0.0f;
    for (int idx = blockIdx.x * blockDim.x + threadIdx.x; idx < n4;
         idx += gridDim.x * blockDim.x) {
        float4 r  = r4[idx];
        float4 im = i4[idx];
        a += r.x * r.x + im.x * im.x;
        a += r.y * r.y + im.y * im.y;
        a += r.z * r.z + im.z * im.z;
        a += r.w * r.w + im.w * im.w;
    }
    float s = block_reduce_256(a, red);
    if (threadIdx.x == 0) atomicAdd(&acc[b * ACC_STRIDE + 10], s);
}

// --------------------------------------------- 50x50 window sums (~51 MB)
// grid = (NREG, BATCH), block = 256; one block owns acc[b][r] -> plain store.
__global__ void region_kernel(const float* __restrict__ ur,
                              const float* __restrict__ ui,
                              const int* __restrict__ sx,
                              const int* __restrict__ sy,
                              const int* __restrict__ sq,
                              float* __restrict__ acc) {
    __shared__ float red[8];
    const int r  = blockIdx.x;
    const int b  = blockIdx.y;
    const int s  = sq[0];
    const int x0 = sx[r];
    const int y0 = sy[r];
    const long base = (long)b * (NMESH * NMESH);
    const int npix = s * s;
    float a = 0.0f;
    for (int w = threadIdx.x; w < npix; w += blockDim.x) {
        int  i   = x0 + w / s;
        int  j   = y0 + w % s;
        long off = base + (long)i * NMESH + j;
        float re = ur[off];
        float im = ui[off];
        a += re * re + im * im;
    }
    float sum = block_reduce_256(a, red);
    if (threadIdx.x == 0) acc[b * ACC_STRIDE + r] = sum;
}

// ------------------------------------------------- finalize (WMMA norm^2)
// 2 blocks x 256 threads = 16 waves; wave gw handles batches gw*16..gw*16+15.
// Lane l and l+16 both own batch gw*16 + (l&15): they supply the two K-pairs
// of one A-row of V_WMMA_F32_16X16X4_F32.  Three chained WMMAs against a
// ones B-matrix reduce the 12 (zero-padded) squared entries over K:
//   D[m][*] = sum_q temp[b_m][q]^2  (the per-sample L2 norm^2).
__global__ void finalize_kernel(const float* __restrict__ acc,
                                float* __restrict__ out) {
    __shared__ float norm2s[8 * 16];
    const int lane = threadIdx.x & 31;
    const int wave = threadIdx.x >> 5;
    const int gw   = blockIdx.x * (blockDim.x >> 5) + wave; // 0..15
    const int m    = lane & 15;
    const int b    = gw * 16 + m;
    const bool lo  = (lane < 16);

    // temp[0..9] = region sums, temp[10] = surrounding, temp[11] = pad 0
    float t0, t1, t2, t3, t4, t5, t6, t7, t8, t9, t10;
    const float* row = acc + b * ACC_STRIDE;
    t0 = row[0]; t1 = row[1]; t2 = row[2]; t3 = row[3]; t4 = row[4];
    t5 = row[5]; t6 = row[6]; t7 = row[7]; t8 = row[8]; t9 = row[9];
    float total = row[10];
    float rsum  = t0 + t1 + t2 + t3 + t4 + t5 + t6 + t7 + t8 + t9;
    t10 = total - rsum;

    // A-matrix K-pairs: lanes 0-15 hold K={0,1}-of-chunk, lanes 16-31 K={2,3}
    v2f a0, a1, a2;
    a0.x = lo ? t0 * t0 : t2 * t2;   // chunk q in {0,1,2,3}
    a0.y = lo ? t1 * t1 : t3 * t3;
    a1.x = lo ? t4 * t4 : t6 * t6;   // chunk q in {4,5,6,7}
    a1.y = lo ? t5 * t5 : t7 * t7;
    a2.x = lo ? t8 * t8 : t10 * t10; // chunk q in {8,9,10,11(=0)}
    a2.y = lo ? t9 * t9 : 0.0f;

    v2f ones; ones.x = 1.0f; ones.y = 1.0f;
    v8f d = {};
    d = __builtin_amdgcn_wmma_f32_16x16x4_f32(false, a0, false, ones,
                                              (short)0, d, false, false);
    d = __builtin_amdgcn_wmma_f32_16x16x4_f32(false, a1, false, ones,
                                              (short)0, d, false, false);
    d = __builtin_amdgcn_wmma_f32_16x16x4_f32(false, a2, false, ones,
                                              (short)0, d, false, false);

    // C/D layout: vgpr v holds row v (lanes 0-15) / row v+8 (lanes 16-31),
    // replicated across columns.  Lane 0 writes rows 0-7, lane 16 rows 8-15.
    if (m == 0) {
        const int rbase = lo ? 0 : 8;
        #pragma unroll
        for (int v = 0; v < 8; ++v) norm2s[wave * 16 + rbase + v] = d[v];
    }
    __syncthreads();

    const float inv = 1.0f / sqrtf(norm2s[wave * 16 + m]);
    if (lo) {
        float* o = out + b * 11;
        o[0]  = t0  * inv; o[1] = t1 * inv; o[2] = t2 * inv; o[3] = t3 * inv;
        o[4]  = t4  * inv; o[5] = t5 * inv; o[6] = t6 * inv; o[7] = t7 * inv;
        o[8]  = t8  * inv; o[9] = t9 * inv; o[10] = t10 * inv;
    }
}

// ------------------------------------------------------------------ launch
extern "C" void kernel_launch(void* const* d_in, const int* in_sizes, int n_in,
                              void* d_out, int out_size, void* d_ws, size_t ws_size,
                              hipStream_t stream) {
    (void)in_sizes; (void)n_in; (void)out_size; (void)ws_size;
    const float* ur = (const float*)d_in[0];
    const float* ui = (const float*)d_in[1];
    const int*   sx = (const int*)d_in[2];
    const int*   sy = (const int*)d_in[3];
    const int*   sq = (const int*)d_in[4];
    float* out = (float*)d_out;
    float* acc = (float*)d_ws; // BATCH*ACC_STRIDE floats = 16 KB

    zero_accum<<<(BATCH * ACC_STRIDE + 255) / 256, 256, 0, stream>>>(acc);
    total_kernel<<<dim3(32, BATCH), 256, 0, stream>>>(ur, ui, acc);
    region_kernel<<<dim3(NREG, BATCH), 256, 0, stream>>>(ur, ui, sx, sy, sq, acc);
    finalize_kernel<<<2, 256, 0, stream>>>(acc, out);
}